// EXPWeightedDILATEloss_35476429865784
// MI455X (gfx1250) — compile-verified
//
#include <hip/hip_runtime.h>

// Exp-time-weighted DILATE loss, B=64, N=512, F=1.
// One forward wavefront DP per batch computes the soft-DTW value AND the
// JVP <dR/dD, Omega> (tangent propagation) in a single sweep — no R-matrix
// storage, no backward pass. One workgroup per batch; 256 threads handle
// 2 rows each so odd->even row dependencies live in registers; only even
// rows are published through a 2-buffer LDS ping-pong. Input row staged via
// CDNA5 async global->LDS (global_load_async_to_lds_b64 + s_wait_asynccnt).

namespace {
constexpr int   kN      = 512;
constexpr int   kB      = 64;
constexpr int   kT      = kN / 2;     // 256 threads, 2 rows per thread (8 waves)
constexpr float kAlpha  = 0.8f;
constexpr float kGamma  = 0.01f;
constexpr float kInvGam = 100.0f;     // 1/gamma
constexpr float kG      = 0.05f;
constexpr float kBig    = 1e8f;
}  // namespace

// softmin_gamma with max-shift; softmax weights reused for the JVP combine.
__device__ __forceinline__ void softmin_jvp(float a, float b, float c,
                                            float ad, float bd, float cd,
                                            float& sm, float& smd) {
  const float m  = fminf(a, fminf(b, c));
  const float ea = __expf((m - a) * kInvGam);
  const float eb = __expf((m - b) * kInvGam);
  const float ec = __expf((m - c) * kInvGam);
  const float z  = ea + eb + ec;                 // >= 1 always (min term -> 1)
  sm  = m - kGamma * __logf(z);
  smd = (ea * ad + eb * bd + ec * cd) / z;
}

__global__ __launch_bounds__(kT) void dilate_softdtw_jvp_kernel(
    const float* __restrict__ inp, const float* __restrict__ tgt,
    float* __restrict__ partials) {
  const int b  = blockIdx.x;
  const int t  = threadIdx.x;         // 0..255
  const int i1 = 2 * t + 1;           // odd DP row: stays in registers
  const int i2 = 2 * t + 2;           // even DP row: published to LDS

  __shared__ float xs[kN];            // input row (the "j" axis)
  __shared__ float vbuf[2][kT + 1];   // even-row values, ping-pong by step parity
  __shared__ float dbuf[2][kT + 1];   // even-row tangents

  // ---- Stage this batch's input row into LDS with the CDNA5 async path. ----
  {
    const float* gp = inp + (size_t)b * kN + 2 * t;
    // Low 32 bits of a flat LDS pointer are the LDS byte offset (ISA 10.2).
    unsigned lds_off = (unsigned)(unsigned long long)(const void*)&xs[2 * t];
    unsigned long long ga = (unsigned long long)(const void*)gp;
    asm volatile("global_load_async_to_lds_b64 %0, %1, off"
                 :: "v"(lds_off), "v"(ga)
                 : "memory");
  }

  // Per-row constants: target samples + normalized exp-time weights.
  const float tv1   = tgt[(size_t)b * kN + 2 * t];
  const float tv2   = tgt[(size_t)b * kN + 2 * t + 1];
  const float wsum  = (__expf(kG * (float)kN) - 1.0f) / (__expf(kG) - 1.0f);
  const float wnorm = (float)kN / wsum;             // mean(w) == 1
  const float w1    = __expf(kG * (float)(2 * t)) * wnorm;
  const float w2    = __expf(kG * (float)(2 * t + 1)) * wnorm;

  // Init the parity-1 buffer (the "r1" diagonal: all BIG) and boundary slot 0
  // (row 0 of R: BIG for j>=1; R[0,0]=0 is carried in thread 0's a1 register).
  vbuf[1][t + 1] = kBig; dbuf[1][t + 1] = 0.0f;
  if (t == 0) {
    vbuf[0][0] = kBig; dbuf[0][0] = 0.0f;
    vbuf[1][0] = kBig; dbuf[1][0] = 0.0f;
  }

  // Register-carried DP state.
  float a1 = (t == 0) ? 0.0f : kBig, ad1 = 0.0f;  // R[i1-1, j1-1]
  float c1 = kBig, cd1 = 0.0f;                    // R[i1,   j1-1] (own prev)
  float h1 = kBig, hd1 = 0.0f;                    // row-i1 output @ step k-1
  float h2 = kBig, hd2 = 0.0f;                    // row-i1 output @ step k-2
  float c2 = kBig, cd2 = 0.0f;                    // R[i2,   j2-1] (own prev)
  float x2 = 0.0f;                                // xs value for row i2
  float lastv = kBig, lastt = 0.0f;

  asm volatile("s_wait_asynccnt 0" ::: "memory");
  __syncthreads();

  // 2N-1 dependent anti-diagonal steps: the serial critical path.
  for (int k = 0; k < 2 * kN - 1; ++k) {
    const int wp = k & 1;             // write parity
    const int rp = wp ^ 1;            // read parity (written at step k-1)
    const int j1 = k + 2 - i1;
    const int j2 = j1 - 1;            // j2(k) == j1(k-1)
    const bool valid1 = (j1 >= 1) && (j1 <= kN);
    const bool valid2 = (j2 >= 1) && (j2 <= kN);

    // Row i1: neighbor row 2t comes from the adjacent thread via LDS.
    const float b1  = vbuf[rp][t];
    const float bd1 = dbuf[rp][t];
    const float x1  = xs[valid1 ? (j1 - 1) : 0];

    float sm1, smd1;
    softmin_jvp(a1, b1, c1, ad1, bd1, cd1, sm1, smd1);
    const float df1  = tv1 - x1;
    const float dij1 = (float)(i1 - j1);
    const float nv1  = valid1 ? (w1 * df1 * df1 + sm1) : kBig;
    const float nt1  = valid1 ? (dij1 * dij1 + smd1) : 0.0f;

    // Row i2: neighbor row i1 entirely from registers (h2 = a, h1 = b).
    float sm2, smd2;
    softmin_jvp(h2, h1, c2, hd2, hd1, cd2, sm2, smd2);
    const float df2  = tv2 - x2;
    const float dij2 = dij1 + 2.0f;   // (i2 - j2) = (i1 - j1) + 2
    const float nv2  = valid2 ? (w2 * df2 * df2 + sm2) : kBig;
    const float nt2  = valid2 ? (dij2 * dij2 + smd2) : 0.0f;

    // Publish the even row for the neighbor thread.
    vbuf[wp][t + 1] = nv2;
    dbuf[wp][t + 1] = nt2;

    // Rotate register state: a(k+1)=b(k); c(k+1)=own output; h-chain shifts.
    a1 = b1;  ad1 = bd1;
    c1 = nv1; cd1 = nt1;
    h2 = h1;  hd2 = hd1;
    h1 = nv1; hd1 = nt1;
    c2 = nv2; cd2 = nt2;
    x2 = x1;
    if (valid2) { lastv = nv2; lastt = nt2; }
    __syncthreads();
  }

  // Thread owning row N computed cell (N,N) on the final diagonal.
  if (t == kT - 1) {
    partials[2 * b]     = lastv;   // soft-DTW value R[N,N]
    partials[2 * b + 1] = lastt;   // JVP: sum(E * Omega)
  }
}

__global__ void dilate_reduce_kernel(const float* __restrict__ partials,
                                     float* __restrict__ out) {
  if (threadIdx.x == 0 && blockIdx.x == 0) {
    float s = 0.0f, st = 0.0f;
    for (int b = 0; b < kB; ++b) {   // fixed order -> deterministic
      s  += partials[2 * b];
      st += partials[2 * b + 1];
    }
    const float invB     = 1.0f / (float)kB;
    const float shape    = s * invB;
    const float temporal = (st * invB) / ((float)kN * (float)kN);
    out[0] = kAlpha * shape + (1.0f - kAlpha) * temporal;
  }
}

extern "C" void kernel_launch(void* const* d_in, const int* in_sizes, int n_in,
                              void* d_out, int out_size, void* d_ws, size_t ws_size,
                              hipStream_t stream) {
  (void)in_sizes; (void)n_in; (void)out_size; (void)ws_size;
  const float* inp = (const float*)d_in[0];   // "input"  [64,512,1] f32
  const float* tgt = (const float*)d_in[1];   // "target" [64,512,1] f32
  float* out      = (float*)d_out;            // scalar f32
  float* partials = (float*)d_ws;             // 2 floats per batch (512 B)

  dilate_softdtw_jvp_kernel<<<kB, kT, 0, stream>>>(inp, tgt, partials);
  dilate_reduce_kernel<<<1, 32, 0, stream>>>(partials, out);
}